// BatchTopKActivationFunction_4587025072459
// MI455X (gfx1250) — compile-verified
//
#include <hip/hip_runtime.h>
#include <stdint.h>

// ----------------- problem constants -----------------
#define K_PER_TOKEN   32
#define THRESH_LR     0.01f
#define F_DIM         16384          // features per token (B=8,S=256,F=16384)

// radix-select split of the 32-bit monotone key: 13 | 13 | 6 bits
#define L1_BITS 13
#define L2_BITS 13
#define L3_BITS 6
#define L1_BINS (1 << L1_BITS)       // 8192
#define L2_BINS (1 << L2_BITS)       // 8192
#define L3_BINS (1 << L3_BITS)       // 64

struct Scal {
  unsigned min_pos_key;   // atomic-min over keys of strictly positive values
  unsigned prefix1, above1;
  unsigned prefix2, above2;
  unsigned kth_key;
  int      tie_budget;
  unsigned finalized;     // 1 => fewer candidates than k_total: select everything
};

// order-preserving float32 -> uint32 key (larger float => larger key)
__device__ __forceinline__ unsigned mono_key(float f) {
  unsigned u = __float_as_uint(f);
  return (u & 0x80000000u) ? ~u : (u | 0x80000000u);
}
__device__ __forceinline__ float key_to_float(unsigned k) {
  return __uint_as_float((k & 0x80000000u) ? (k & 0x7FFFFFFFu) : ~k);
}

// ----------------- CDNA5 TDM + tensorcnt helpers -----------------
#if __has_builtin(__builtin_amdgcn_tensor_load_to_lds)
#define HAVE_TDM 1
typedef unsigned u32x4 __attribute__((ext_vector_type(4)));
typedef int      i32x4 __attribute__((ext_vector_type(4)));
typedef int      i32x8 __attribute__((ext_vector_type(8)));

// 1-D DMA of nelem fp32 from global to LDS via the Tensor Data Mover.
// D# built per CDNA5 ISA (group0 128b, group1 256b).
// This toolchain (clang-23 / therock-10.0) exposes the 6-arg builtin:
//   (uint32x4 g0, int32x8 g1, int32x4, int32x4, int32x8, i32 cpol)
__device__ __forceinline__ void tdm_load_row(const float* gsrc, float* lds_dst, int nelem) {
  uint64_t ga = (uint64_t)(uintptr_t)gsrc;
  // flat shared address carries the LDS byte offset in its low 32 bits (ISA 10.2)
  unsigned lo = (unsigned)(uintptr_t)lds_dst;
  u32x4 g0;
  i32x8 g1;
  i32x4 g2 = {0, 0, 0, 0};
  i32x4 g3 = {0, 0, 0, 0};
  i32x8 g4 = {0, 0, 0, 0, 0, 0, 0, 0};
  g0[0] = 1u;                                                   // count=1 valid D#
  g0[1] = lo;                                                   // lds_addr (bytes)
  g0[2] = (unsigned)(ga & 0xFFFFFFFFu);                         // global_addr[31:0]
  g0[3] = (unsigned)((ga >> 32) & 0x01FFFFFFu) | 0x80000000u;   // [56:32] | type=2
  g1[0] = 2 << 16;                                              // data_size = 4 bytes
  g1[1] = (int)((unsigned)(nelem & 0xFFFF) << 16);              // tensor_dim0[15:0]
  g1[2] = (int)(((unsigned)nelem >> 16) | (1u << 16));          // tdim0[31:16], tensor_dim1=1
  g1[3] = (int)((unsigned)(nelem & 0xFFFF) << 16);              // tile_dim0
  g1[4] = 1;                                                    // tile_dim1=1, tile_dim2=0
  g1[5] = nelem;                                                // tensor_dim0_stride[31:0]
  g1[6] = 0;
  g1[7] = 0;
  __builtin_amdgcn_tensor_load_to_lds(g0, g1, g2, g3, g4, 0);
}
#else
#define HAVE_TDM 0
#endif

__device__ __forceinline__ void wait_tensorcnt0() {
#if __has_builtin(__builtin_amdgcn_s_wait_tensorcnt)
  __builtin_amdgcn_s_wait_tensorcnt(0);
#else
  asm volatile("s_wait_tensorcnt 0x0" ::: "memory");
#endif
}
__device__ __forceinline__ void wait_tensorcnt1() {
#if __has_builtin(__builtin_amdgcn_s_wait_tensorcnt)
  __builtin_amdgcn_s_wait_tensorcnt(1);
#else
  asm volatile("s_wait_tensorcnt 0x1" ::: "memory");
#endif
}

// ----------------- kernels -----------------

__global__ void init_ws(unsigned* ws) {
  const int nhist = L1_BINS + L2_BINS + L3_BINS;
  int i = blockIdx.x * blockDim.x + threadIdx.x;
  if (i < nhist) ws[i] = 0u;
  if (i == 0) {
    Scal* sc = (Scal*)(ws + nhist);
    sc->min_pos_key = 0xFFFFFFFFu;
    sc->prefix1 = 0u; sc->above1 = 0u;
    sc->prefix2 = 0u; sc->above2 = 0u;
    sc->kth_key = 0u; sc->tie_budget = 0; sc->finalized = 0u;
  }
}

// Pass 1: per-WG, double-buffered TDM staging of token rows (64 KiB each) into
// LDS; 8 wave32 waves histogram the top 13 key bits via ds_add into a 32 KiB
// LDS histogram while the TDM DMAs the next row. One global-atomic flush at
// the end. Masked tokens are skipped entirely (they can never be selected
// except in the degenerate case handled by select_big's "finalized" path).
__global__ void __launch_bounds__(256)
pass1_hist(const float* __restrict__ x, const int* __restrict__ mask,
           unsigned* __restrict__ g_hist1, Scal* __restrict__ sc, int n_tokens) {
  __shared__ __align__(16) float buf[2][F_DIM];     // 2 x 64 KiB staging
  __shared__ unsigned hist[L1_BINS];                // 32 KiB histogram
  __shared__ unsigned wg_minpos;
  const int tid  = threadIdx.x;
  const int step = gridDim.x;

  for (int b = tid; b < L1_BINS; b += 256) hist[b] = 0u;
  if (tid == 0) wg_minpos = 0xFFFFFFFFu;
  __syncthreads();

  unsigned my_minpos = 0xFFFFFFFFu;

  // first active (unmasked) token for this block; scan is uniform/scalar
  int cur = blockIdx.x;
  while (cur < n_tokens && mask[cur] == 0) cur += step;

  int pb = 0;
#if HAVE_TDM
  if (cur < n_tokens && tid == 0) tdm_load_row(x + (size_t)cur * F_DIM, buf[0], F_DIM);
#endif

  while (cur < n_tokens) {
    // locate next active token and prefetch it into the alternate buffer
    int nxt = cur + step;
    while (nxt < n_tokens && mask[nxt] == 0) nxt += step;
#if HAVE_TDM
    if (nxt < n_tokens) {
      if (tid == 0) tdm_load_row(x + (size_t)nxt * F_DIM, buf[pb ^ 1], F_DIM);
      wait_tensorcnt1();                 // TDM ops retire in order: oldest (cur) done
    } else {
      wait_tensorcnt0();
    }
    __syncthreads();
#else
    for (int i = tid * 4; i < F_DIM; i += 256 * 4)
      *(float4*)&buf[pb][i] = *(const float4*)&x[(size_t)cur * F_DIM + i];
    __syncthreads();
#endif
    for (int i = tid * 4; i < F_DIM; i += 256 * 4) {
      float4 v = *(const float4*)&buf[pb][i];
      const float vv[4] = {v.x, v.y, v.z, v.w};
#pragma unroll
      for (int c = 0; c < 4; ++c) {
        unsigned key = mono_key(vv[c]);
        atomicAdd(&hist[key >> (32 - L1_BITS)], 1u);
        if (key > 0x80000000u && key < my_minpos) my_minpos = key;  // strictly positive
      }
    }
    __syncthreads();                     // readers done before this buffer is re-DMA'd
    cur = nxt;
    pb ^= 1;
  }

  atomicMin(&wg_minpos, my_minpos);
  __syncthreads();
  for (int b = tid; b < L1_BINS; b += 256) {
    unsigned c = hist[b];
    if (c) atomicAdd(&g_hist1[b], c);
  }
  if (tid == 0 && wg_minpos != 0xFFFFFFFFu) atomicMin(&sc->min_pos_key, wg_minpos);
}

// Top-down scan of an 8192-bin histogram; finds the bin containing the k-th key.
__global__ void __launch_bounds__(256)
select_big(const unsigned* __restrict__ hist, Scal* __restrict__ sc,
           int level, unsigned k_total) {
  __shared__ unsigned segsum[256];
  __shared__ unsigned s_seg, s_rem;
  const int BPT = L1_BINS / 256;                    // 32 bins per thread
  if (level == 2 && sc->finalized) return;          // uniform
  const unsigned rem0 = (level == 1) ? k_total : (k_total - sc->above1);
  const int tid = threadIdx.x;
  const int hi = L1_BINS - 1 - tid * BPT;           // segments ordered from the top

  unsigned sum = 0;
  for (int j = 0; j < BPT; ++j) sum += hist[hi - j];
  segsum[tid] = sum;
  __syncthreads();

  if (tid == 0) {
    unsigned acc = 0, remAt = 0; int seg = -1;
    for (int s = 0; s < 256; ++s) {
      if (acc + segsum[s] >= rem0) { seg = s; remAt = rem0 - acc; break; }
      acc += segsum[s];
    }
    if (seg < 0) {                                  // total < k_total: select all
      sc->finalized = 1u; sc->kth_key = 0u; sc->tie_budget = 0;
      sc->prefix1 = 0xFFFFFFFFu; sc->prefix2 = 0xFFFFFFFFu;
      s_seg = 0xFFFFFFFFu;
    } else { s_seg = (unsigned)seg; s_rem = remAt; }
  }
  __syncthreads();
  if (s_seg == 0xFFFFFFFFu) return;

  if (tid == (int)s_seg) {
    unsigned acc = 0;
    for (int j = 0; j < BPT; ++j) {
      unsigned bin = (unsigned)(hi - j);
      unsigned c = hist[hi - j];
      if (acc + c >= s_rem) {
        unsigned aboveG = (rem0 - s_rem) + acc;     // strictly above chosen bin
        if (level == 1) { sc->prefix1 = bin; sc->above1 = aboveG; }
        else {
          sc->prefix2 = (sc->prefix1 << L2_BITS) | bin;
          sc->above2  = sc->above1 + aboveG;
        }
        break;
      }
      acc += c;
    }
  }
}

// Pass 2: refine next 13 bits among elements whose top-13 matches prefix1.
// x is L2-resident after pass 1 (128 MiB < 192 MiB L2), so this is cheap.
__global__ void __launch_bounds__(256)
pass2_hist(const float4* __restrict__ x4, const int* __restrict__ mask,
           unsigned* __restrict__ g_hist2, const Scal* __restrict__ sc, long long n4) {
  const unsigned p1 = sc->prefix1;
  if (p1 >= (unsigned)L1_BINS) return;              // finalized / sentinel
  const long long stride = (long long)gridDim.x * blockDim.x;
  for (long long i = (long long)blockIdx.x * blockDim.x + threadIdx.x; i < n4; i += stride) {
    long long e = i * 4;
    if (!mask[e / F_DIM]) continue;
    float4 v = x4[i];
    const float vv[4] = {v.x, v.y, v.z, v.w};
#pragma unroll
    for (int c = 0; c < 4; ++c) {
      unsigned key = mono_key(vv[c]);
      if ((key >> (32 - L1_BITS)) == p1)
        atomicAdd(&g_hist2[(key >> L3_BITS) & (L2_BINS - 1)], 1u);
    }
  }
}

// Pass 3: final 6 bits among elements whose top-26 matches prefix2.
__global__ void __launch_bounds__(256)
pass3_hist(const float4* __restrict__ x4, const int* __restrict__ mask,
           unsigned* __restrict__ g_hist3, const Scal* __restrict__ sc, long long n4) {
  const unsigned p2 = sc->prefix2;
  if (p2 >= (1u << (L1_BITS + L2_BITS))) return;    // finalized / sentinel
  const long long stride = (long long)gridDim.x * blockDim.x;
  for (long long i = (long long)blockIdx.x * blockDim.x + threadIdx.x; i < n4; i += stride) {
    long long e = i * 4;
    if (!mask[e / F_DIM]) continue;
    float4 v = x4[i];
    const float vv[4] = {v.x, v.y, v.z, v.w};
#pragma unroll
    for (int c = 0; c < 4; ++c) {
      unsigned key = mono_key(vv[c]);
      if ((key >> L3_BITS) == p2)
        atomicAdd(&g_hist3[key & (L3_BINS - 1)], 1u);
    }
  }
}

// Finalize k-th key + tie budget; compute EMA threshold and write to out[n].
__global__ void select3_thr(const unsigned* __restrict__ h3, Scal* __restrict__ sc,
                            const float* __restrict__ thr_in, float* __restrict__ out,
                            long long n, unsigned k_total) {
  if (threadIdx.x != 0 || blockIdx.x != 0) return;
  if (!sc->finalized) {
    unsigned rem = k_total - sc->above2;
    unsigned acc = 0;
    for (int b = L3_BINS - 1; b >= 0; --b) {
      unsigned c = h3[b];
      if (acc + c >= rem) {
        sc->kth_key   = (sc->prefix2 << L3_BITS) | (unsigned)b;
        sc->tie_budget = (int)(rem - acc);          // #elements equal to kth to accept
        break;
      }
      acc += c;
    }
  }
  // EMA: any positive selected <=> any positive masked value exists
  float thr = thr_in[0];
  if (sc->min_pos_key != 0xFFFFFFFFu) {
    unsigned kk = sc->kth_key;
    float mp = (kk > 0x80000000u) ? key_to_float(kk)            // kth value itself positive
                                  : key_to_float(sc->min_pos_key); // all positives selected
    thr = (1.0f - THRESH_LR) * thr + THRESH_LR * mp;
  }
  out[n] = thr;
}

// Pass 4: emit relu(x) where selected, else 0. Ties at the exact k-th key are
// claimed via an atomic budget (only needed when kth value is positive).
__global__ void __launch_bounds__(256)
pass4_out(const float4* __restrict__ x4, const int* __restrict__ mask,
          float4* __restrict__ out4, Scal* __restrict__ sc, long long n4) {
  const unsigned kth = sc->kth_key;
  const bool tie_pos = kth > 0x80000000u;           // kth value strictly positive
  const long long stride = (long long)gridDim.x * blockDim.x;
  for (long long i = (long long)blockIdx.x * blockDim.x + threadIdx.x; i < n4; i += stride) {
    long long e = i * 4;
    float4 o = {0.f, 0.f, 0.f, 0.f};
    if (mask[e / F_DIM]) {
      float4 v = x4[i];
      float vv[4] = {v.x, v.y, v.z, v.w};
      float oo[4] = {0.f, 0.f, 0.f, 0.f};
#pragma unroll
      for (int c = 0; c < 4; ++c) {
        unsigned key = mono_key(vv[c]);
        if (key > kth) {
          if (vv[c] > 0.f) oo[c] = vv[c];           // selected; relu
        } else if (key == kth && tie_pos) {
          int old = atomicAdd(&sc->tie_budget, -1);
          if (old > 0) oo[c] = vv[c];               // positive by construction
        }
      }
      o.x = oo[0]; o.y = oo[1]; o.z = oo[2]; o.w = oo[3];
    }
    out4[i] = o;
  }
}

// ----------------- launcher -----------------
extern "C" void kernel_launch(void* const* d_in, const int* in_sizes, int n_in,
                              void* d_out, int out_size, void* d_ws, size_t ws_size,
                              hipStream_t stream) {
  const float* x      = (const float*)d_in[0];
  const int*   mask   = (const int*)d_in[1];
  const float* thr_in = (const float*)d_in[2];
  float*       out    = (float*)d_out;

  const long long n        = (long long)in_sizes[0];   // B*S*F elements
  const int       n_tokens = in_sizes[1];              // B*S
  const long long n4       = n / 4;
  const unsigned  k_total  = (unsigned)(K_PER_TOKEN * n_tokens);

  unsigned* ws = (unsigned*)d_ws;
  unsigned* h1 = ws;
  unsigned* h2 = ws + L1_BINS;
  unsigned* h3 = ws + L1_BINS + L2_BINS;
  Scal*     sc = (Scal*)(ws + L1_BINS + L2_BINS + L3_BINS);

  const int nhist = L1_BINS + L2_BINS + L3_BINS;
  init_ws<<<(nhist + 255) / 256, 256, 0, stream>>>(ws);
  pass1_hist<<<256, 256, 0, stream>>>(x, mask, h1, sc, n_tokens);
  select_big<<<1, 256, 0, stream>>>(h1, sc, 1, k_total);
  pass2_hist<<<1024, 256, 0, stream>>>((const float4*)x, mask, h2, sc, n4);
  select_big<<<1, 256, 0, stream>>>(h2, sc, 2, k_total);
  pass3_hist<<<1024, 256, 0, stream>>>((const float4*)x, mask, h3, sc, n4);
  select3_thr<<<1, 32, 0, stream>>>(h3, sc, thr_in, out, n, k_total);
  pass4_out<<<2048, 256, 0, stream>>>((const float4*)x, mask, (float4*)out, sc, n4);
}